// NequIPEnergyModel_51908974740079
// MI455X (gfx1250) — compile-verified
//
#include <hip/hip_runtime.h>
#include <math.h>

#define N_NODES 20000
#define N_EDGES 320000
#define G_GRAPHS 16

typedef __attribute__((ext_vector_type(2))) float v2f;
typedef __attribute__((ext_vector_type(8))) float v8f;

// silu with hardware approximate reciprocal (v_rcp_f32): avoids the IEEE
// div expansion (div_scale/div_fmas chains) on the WMMA->LDS critical path.
__device__ __forceinline__ float silu_f(float x) {
  return x * __builtin_amdgcn_rcpf(1.0f + __expf(-x));
}

__device__ __forceinline__ v8f wmma_f32x4(v2f a, v2f b, v8f c) {
  // 16x16x4 fp32 WMMA: D = A(16x4) * B(4x16) + C(16x16)
  return __builtin_amdgcn_wmma_f32_16x16x4_f32(false, a, false, b, (short)0, c,
                                               false, false);
}

// ---------------------------------------------------------------------------
// s = nodes @ W_embed / sqrt(NEL)
// ---------------------------------------------------------------------------
__global__ void embed_kernel(const float* __restrict__ nodes,
                             const float* __restrict__ Wemb,
                             float* __restrict__ s) {
  const int idx = blockIdx.x * blockDim.x + threadIdx.x;
  if (idx >= N_NODES * 64) return;
  const int n = idx >> 6, o = idx & 63;
  float acc = 0.f;
#pragma unroll
  for (int e = 0; e < 4; ++e) acc += nodes[n * 4 + e] * Wemb[e * 64 + o];
  s[idx] = acc * 0.5f;
}

// ---------------------------------------------------------------------------
// Per-node pre-pass: sc_s, sc_v, s1, v1 (one block of 128 threads per node)
// ---------------------------------------------------------------------------
__global__ __launch_bounds__(128) void node_pre_kernel(
    const float* __restrict__ s, const float* __restrict__ v,
    const float* __restrict__ nodes, const float* __restrict__ Wscs,
    const float* __restrict__ Wscv, const float* __restrict__ W1s,
    const float* __restrict__ W1v, float* __restrict__ sc_s,
    float* __restrict__ sc_v, float* __restrict__ s1, float* __restrict__ v1) {
  const int n = blockIdx.x;
  const int tid = threadIdx.x;
  __shared__ float ls[64];
  __shared__ float lv[96];
  __shared__ float lat[4];
  if (tid < 64) ls[tid] = s[n * 64 + tid];
  if (tid < 96) lv[tid] = v[n * 96 + tid];
  if (tid < 4) lat[tid] = nodes[n * 4 + tid];
  __syncthreads();
  // s1 = s @ W1s / sqrt(64)
  for (int o = tid; o < 64; o += 128) {
    float acc = 0.f;
#pragma unroll 8
    for (int i = 0; i < 64; ++i) acc += ls[i] * W1s[i * 64 + o];
    s1[n * 64 + o] = acc * 0.125f;
  }
  // sc_s = einsum('i,e,ieo->o') / sqrt(256)
  for (int o = tid; o < 96; o += 128) {
    float acc = 0.f;
    for (int i = 0; i < 64; ++i) {
      const float si = ls[i];
#pragma unroll
      for (int e = 0; e < 4; ++e) acc += si * lat[e] * Wscs[(i * 4 + e) * 96 + o];
    }
    sc_s[n * 96 + o] = acc * 0.0625f;
  }
  // v1 = einsum('ic,io->oc') / sqrt(32)
  for (int idx = tid; idx < 96; idx += 128) {
    const int o = idx / 3, c = idx % 3;
    float acc = 0.f;
#pragma unroll 8
    for (int i = 0; i < 32; ++i) acc += lv[i * 3 + c] * W1v[i * 32 + o];
    v1[n * 96 + idx] = acc * 0.17677670f;
  }
  // sc_v = einsum('ic,e,ieo->oc') / sqrt(128)
  for (int idx = tid; idx < 96; idx += 128) {
    const int o = idx / 3, c = idx % 3;
    float acc = 0.f;
    for (int i = 0; i < 32; ++i) {
      const float vi = lv[i * 3 + c];
#pragma unroll
      for (int e = 0; e < 4; ++e) acc += vi * lat[e] * Wscv[(i * 4 + e) * 32 + o];
    }
    sc_v[n * 96 + idx] = acc * 0.08838835f;
  }
}

// ---------------------------------------------------------------------------
// Fused edge kernel: bessel embedding -> 3-layer radial MLP via fp32 WMMA
// (16 edges per wave) -> message construction -> atomic scatter (segment sum).
// Never materializes the E x 192 radial weights to HBM.
// ---------------------------------------------------------------------------
__global__ __launch_bounds__(128) void edge_kernel(
    const float* __restrict__ dR, const int* __restrict__ src,
    const int* __restrict__ dst, const float* __restrict__ s1g,
    const float* __restrict__ v1g, const float* __restrict__ R0,
    const float* __restrict__ R1, const float* __restrict__ R2,
    float* __restrict__ a_s, float* __restrict__ a_v) {
  __shared__ float shH1[4][16][64];   // silu(remb@R0) staged row-major
  __shared__ float shH2[4][16][64];   // silu(h1@R1)
  __shared__ float shRemb[4][16][8];  // bessel features (A matrix of GEMM1)
  __shared__ float shSh1[4][16][3];   // sqrt(3)*unit vector
  __shared__ int shSrc[4][16];
  __shared__ int shDst[4][16];

  const int wave = threadIdx.x >> 5;
  const int lane = threadIdx.x & 31;
  const int e0 = (blockIdx.x * 4 + wave) * 16;

  if (lane < 16) {
    const int e = e0 + lane;
    const float x = dR[e * 3 + 0];
    const float y = dR[e * 3 + 1];
    const float z = dR[e * 3 + 2];
    const float r = sqrtf(x * x + y * y + z * z);
    const float rs = fmaxf(r, 1e-6f);
    const float ri = 1.0f / rs;
    shSh1[wave][lane][0] = 1.7320508f * x * ri;
    shSh1[wave][lane][1] = 1.7320508f * y * ri;
    shSh1[wave][lane][2] = 1.7320508f * z * ri;
    shSrc[wave][lane] = src[e];
    shDst[wave][lane] = dst[e];
    float env;
    if (r < 3.5f)      env = 1.0f;
    else if (r > 4.0f) env = 0.0f;
    else               env = 0.5f * (cosf(6.2831853f * (r - 3.5f)) + 1.0f);
    const float base = 0.70710678f * ri * env;  // sqrt(2/RMAX)/r_safe * env
#pragma unroll
    for (int nb = 0; nb < 8; ++nb)
      shRemb[wave][lane][nb] = base * sinf((float)(nb + 1) * 0.78539816f * rs);
  }
  __syncthreads();

  // fp32 WMMA fragment geometry (ISA 7.12.2):
  //  A (16x4): lane<16 holds rows M=lane, K={0,1}; lane>=16 holds K={2,3}
  //  B (4x16): lane<16 holds cols N=lane, K={0,1}; lane>=16 holds K={2,3}
  //  C/D     : VGPR j holds row M=j (lanes<16) or M=j+8 (lanes>=16), N=lane%16
  const int m = lane & 15;
  const int koff = (lane < 16) ? 0 : 2;
  const int mbase = (lane < 16) ? 0 : 8;

  // ---- GEMM1: h1 = silu(remb(16x8) @ R0(8x64) / sqrt(8))
  for (int nt = 0; nt < 4; ++nt) {
    v8f acc = {};
#pragma unroll
    for (int kc = 0; kc < 2; ++kc) {
      const int kr = kc * 4 + koff;
      v2f a, b;
      a.x = shRemb[wave][m][kr];
      a.y = shRemb[wave][m][kr + 1];
      b.x = R0[kr * 64 + nt * 16 + m];
      b.y = R0[(kr + 1) * 64 + nt * 16 + m];
      acc = wmma_f32x4(a, b, acc);
    }
#pragma unroll
    for (int j = 0; j < 8; ++j)
      shH1[wave][j + mbase][nt * 16 + m] = silu_f(acc[j] * 0.35355339f);
  }
  __syncthreads();

  // ---- GEMM2: h2 = silu(h1(16x64) @ R1(64x64) / 8)
  for (int nt = 0; nt < 4; ++nt) {
    v8f acc = {};
#pragma unroll
    for (int kc = 0; kc < 16; ++kc) {
      const int kr = kc * 4 + koff;
      v2f a, b;
      a.x = shH1[wave][m][kr];
      a.y = shH1[wave][m][kr + 1];
      b.x = R1[kr * 64 + nt * 16 + m];
      b.y = R1[(kr + 1) * 64 + nt * 16 + m];
      acc = wmma_f32x4(a, b, acc);
    }
#pragma unroll
    for (int j = 0; j < 8; ++j)
      shH2[wave][j + mbase][nt * 16 + m] = silu_f(acc[j] * 0.125f);
  }
  __syncthreads();

  // ---- GEMM3: w = h2(16x64) @ R2(64x192) / 8, fused message scatter.
  // Tile nt (16 w-channels) lies entirely in one region: w1|w2|w3|w4.
  for (int nt = 0; nt < 12; ++nt) {
    v8f acc = {};
#pragma unroll
    for (int kc = 0; kc < 16; ++kc) {
      const int kr = kc * 4 + koff;
      v2f a, b;
      a.x = shH2[wave][m][kr];
      a.y = shH2[wave][m][kr + 1];
      b.x = R2[kr * 192 + nt * 16 + m];
      b.y = R2[(kr + 1) * 192 + nt * 16 + m];
      acc = wmma_f32x4(a, b, acc);
    }
    const int wch = nt * 16 + m;  // region-uniform across the wave per nt
#pragma unroll
    for (int j = 0; j < 8; ++j) {
      const int el = j + mbase;  // edge within tile
      const float wv = acc[j] * 0.125f;
      const int d = shDst[wave][el];
      const int si = shSrc[wave][el];
      if (wch < 64) {  // m_s[:, 0:64] = w1 * se   (sh0 == 1), /sqrt(NNB)
        atomicAdd(&a_s[d * 96 + wch], wv * s1g[si * 64 + wch] * 0.25f);
      } else if (wch < 128) {  // m_v[:, 0:64, c] = (w2*se)*sh1[c]
        const int i = wch - 64;
        const float t = wv * s1g[si * 64 + i] * 0.25f;
        atomicAdd(&a_v[d * 288 + i * 3 + 0], t * shSh1[wave][el][0]);
        atomicAdd(&a_v[d * 288 + i * 3 + 1], t * shSh1[wave][el][1]);
        atomicAdd(&a_v[d * 288 + i * 3 + 2], t * shSh1[wave][el][2]);
      } else if (wch < 160) {  // m_v[:, 64:96, c] = w3 * ve[:, i, c]
        const int i = wch - 128;
        const float t = wv * 0.25f;
        atomicAdd(&a_v[d * 288 + (64 + i) * 3 + 0], t * v1g[si * 96 + i * 3 + 0]);
        atomicAdd(&a_v[d * 288 + (64 + i) * 3 + 1], t * v1g[si * 96 + i * 3 + 1]);
        atomicAdd(&a_v[d * 288 + (64 + i) * 3 + 2], t * v1g[si * 96 + i * 3 + 2]);
      } else {  // m_s[:, 64:96] = w4 * dot(ve, sh1)/sqrt(3)
        const int i = wch - 160;
        const float dot = v1g[si * 96 + i * 3 + 0] * shSh1[wave][el][0] +
                          v1g[si * 96 + i * 3 + 1] * shSh1[wave][el][1] +
                          v1g[si * 96 + i * 3 + 2] * shSh1[wave][el][2];
        atomicAdd(&a_s[d * 96 + 64 + i], wv * dot * 0.14433757f);
      }
    }
  }
}

// ---------------------------------------------------------------------------
// Node update: o_s = a_s@W2s/sqrt(96)+sc_s ; o_v = a_v@W2v/sqrt(96)+sc_v ;
// s <- silu(o_s[:64]) ; v <- o_v * silu(o_s[64:])
// ---------------------------------------------------------------------------
__global__ __launch_bounds__(128) void node_update_kernel(
    const float* __restrict__ a_s, const float* __restrict__ a_v,
    const float* __restrict__ sc_s, const float* __restrict__ sc_v,
    const float* __restrict__ W2s, const float* __restrict__ W2v,
    float* __restrict__ s, float* __restrict__ v) {
  const int n = blockIdx.x;
  const int tid = threadIdx.x;
  __shared__ float las[96];
  __shared__ float lav[288];
  __shared__ float gate[32];
  if (tid < 96) las[tid] = a_s[n * 96 + tid];
  for (int i = tid; i < 288; i += 128) lav[i] = a_v[n * 288 + i];
  __syncthreads();
  for (int o = tid; o < 96; o += 128) {
    float acc = 0.f;
    for (int i = 0; i < 96; ++i) acc += las[i] * W2s[i * 96 + o];
    const float os = acc * 0.10206207f + sc_s[n * 96 + o];
    if (o < 64) s[n * 64 + o] = silu_f(os);
    else        gate[o - 64] = silu_f(os);
  }
  __syncthreads();
  for (int idx = tid; idx < 96; idx += 128) {
    const int o = idx / 3, c = idx % 3;
    float acc = 0.f;
    for (int i = 0; i < 96; ++i) acc += lav[i * 3 + c] * W2v[i * 32 + o];
    const float ov = acc * 0.10206207f + sc_v[n * 96 + idx];
    v[n * 96 + idx] = ov * gate[o];
  }
}

// ---------------------------------------------------------------------------
// Readout: per-node energy head (atomic into per-graph sum) + magnetization
// ---------------------------------------------------------------------------
__global__ void readout_kernel(
    const float* __restrict__ s, const float* __restrict__ nodes,
    const int* __restrict__ n_node, const float* __restrict__ Wen1,
    const float* __restrict__ Wen2, const float* __restrict__ Wmag1,
    const float* __restrict__ Wmag2, const float* __restrict__ scale_occ,
    const float* __restrict__ shift_occ, const float* __restrict__ escale,
    const float* __restrict__ eshift, float* __restrict__ out) {
  const int n = blockIdx.x * blockDim.x + threadIdx.x;
  if (n >= N_NODES) return;
  float he[32], hm[32];
#pragma unroll
  for (int k = 0; k < 32; ++k) { he[k] = 0.f; hm[k] = 0.f; }
  for (int i = 0; i < 64; ++i) {
    const float si = s[n * 64 + i];
#pragma unroll
    for (int k = 0; k < 32; ++k) {
      he[k] += si * Wen1[i * 32 + k];
      hm[k] += si * Wmag1[i * 32 + k];
    }
  }
  float ed = 0.f;
#pragma unroll
  for (int k = 0; k < 32; ++k) ed += he[k] * Wen2[k];
  float atomic = ed * 0.02209709f;  // (1/sqrt(64)) * (1/sqrt(32))
  atomic = escale[0] * atomic + eshift[0];
  int g = G_GRAPHS - 1, cum = 0;
  for (int gg = 0; gg < G_GRAPHS; ++gg) {
    cum += n_node[gg];
    if (n < cum) { g = gg; break; }
  }
  atomicAdd(&out[g], atomic);
#pragma unroll
  for (int k2 = 0; k2 < 2; ++k2) {
    float md = 0.f;
#pragma unroll
    for (int k = 0; k < 32; ++k) md += hm[k] * Wmag2[k * 2 + k2];
    md *= 0.02209709f;
    float sc = 0.f, sh = 0.f;
#pragma unroll
    for (int j = 0; j < 3; ++j) {
      sc += nodes[n * 4 + j] * scale_occ[j * 2 + k2];
      sh += nodes[n * 4 + j] * shift_occ[j * 2 + k2];
    }
    out[16 + n * 2 + k2] = sc * md + sh;
  }
}

// ---------------------------------------------------------------------------
extern "C" void kernel_launch(void* const* d_in, const int* in_sizes, int n_in,
                              void* d_out, int out_size, void* d_ws,
                              size_t ws_size, hipStream_t stream) {
  (void)in_sizes; (void)n_in; (void)out_size; (void)ws_size;
  const float* nodes     = (const float*)d_in[0];
  const float* dR        = (const float*)d_in[1];
  const int*   senders   = (const int*)d_in[2];
  const int*   receivers = (const int*)d_in[3];
  const int*   n_node    = (const int*)d_in[4];
  const float* W_embed   = (const float*)d_in[5];
  const float* W_sc_s    = (const float*)d_in[6];
  const float* W_sc_v    = (const float*)d_in[7];
  const float* W_l1_s    = (const float*)d_in[8];
  const float* W_l1_v    = (const float*)d_in[9];
  const float* Wr0       = (const float*)d_in[10];
  const float* Wr1       = (const float*)d_in[11];
  const float* Wr2       = (const float*)d_in[12];
  const float* W_l2_s    = (const float*)d_in[13];
  const float* W_l2_v    = (const float*)d_in[14];
  const float* W_en1     = (const float*)d_in[15];
  const float* W_en2     = (const float*)d_in[16];
  const float* W_mag1    = (const float*)d_in[17];
  const float* W_mag2    = (const float*)d_in[18];
  const float* scale_occ = (const float*)d_in[19];
  const float* shift_occ = (const float*)d_in[20];
  const float* escale    = (const float*)d_in[21];
  const float* eshift    = (const float*)d_in[22];

  float* ws  = (float*)d_ws;
  float* s   = ws;                     // N*64
  float* s1  = s + N_NODES * 64;       // N*64
  float* v   = s1 + N_NODES * 64;      // N*96
  float* v1  = v + N_NODES * 96;       // N*96
  float* scs = v1 + N_NODES * 96;      // N*96
  float* scv = scs + N_NODES * 96;     // N*96
  float* as_ = scv + N_NODES * 96;     // N*96
  float* av_ = as_ + N_NODES * 96;     // N*288  (total ~72 MB)

  hipMemsetAsync(v, 0, N_NODES * 96 * sizeof(float), stream);
  embed_kernel<<<(N_NODES * 64 + 255) / 256, 256, 0, stream>>>(nodes, W_embed, s);

  for (int l = 0; l < 2; ++l) {
    node_pre_kernel<<<N_NODES, 128, 0, stream>>>(
        s, v, nodes, W_sc_s + l * 64 * 4 * 96, W_sc_v + l * 32 * 4 * 32,
        W_l1_s + l * 64 * 64, W_l1_v + l * 32 * 32, scs, scv, s1, v1);
    hipMemsetAsync(as_, 0, N_NODES * 96 * sizeof(float), stream);
    hipMemsetAsync(av_, 0, N_NODES * 288 * sizeof(float), stream);
    edge_kernel<<<N_EDGES / 64, 128, 0, stream>>>(
        dR, senders, receivers, s1, v1, Wr0 + l * 8 * 64, Wr1 + l * 64 * 64,
        Wr2 + l * 64 * 192, as_, av_);
    node_update_kernel<<<N_NODES, 128, 0, stream>>>(
        as_, av_, scs, scv, W_l2_s + l * 96 * 96, W_l2_v + l * 96 * 32, s, v);
  }

  hipMemsetAsync(d_out, 0, G_GRAPHS * sizeof(float), stream);
  readout_kernel<<<(N_NODES + 127) / 128, 128, 0, stream>>>(
      s, nodes, n_node, W_en1, W_en2, W_mag1, W_mag2, scale_occ, shift_occ,
      escale, eshift, (float*)d_out);
}